// GAT_64441689309932
// MI455X (gfx1250) — compile-verified
//
#include <hip/hip_runtime.h>
#include <hip/hip_bf16.h>
#include <math.h>

typedef __attribute__((ext_vector_type(2))) float v2f;
typedef __attribute__((ext_vector_type(8))) float v8f;

#define HEADS 3
#define CH 64
#define HC 192      // HEADS*CH
#define FIN 256
#define OUTF 40
#define NEG_SLOPE 0.2f

__device__ __forceinline__ float lrelu(float v) {
    return v > 0.f ? v : NEG_SLOPE * v;
}

// ---------------- Kernel 1: h = x @ W1 via V_WMMA_F32_16X16X4_F32 ----------------
// One wave computes one 16x16 tile of h. M tiles = N/16 (exact), N tiles = 12.
__global__ void gat_gemm1_wmma(const float* __restrict__ x,
                               const float* __restrict__ W1,
                               float* __restrict__ h,
                               int n_tiles_total, int m_tiles_unused) {
    int wave = (int)((blockIdx.x * blockDim.x + threadIdx.x) >> 5);
    int lane = threadIdx.x & 31;
    if (wave < n_tiles_total) {
        const int NT = HC / 16;             // 12
        int mt = wave / NT;
        int nt = wave - mt * NT;
        int row0 = mt * 16, col0 = nt * 16;
        int idx  = lane & 15;
        int half = lane >> 4;               // selects K pair {0,1} vs {2,3}

        const float* xr = x  + (size_t)(row0 + idx) * FIN + 2 * half;   // A: row=row0+idx
        const float* wc = W1 + (size_t)(2 * half) * HC + (col0 + idx);  // B: col=col0+idx

        v8f acc = {};
        #pragma unroll 4
        for (int k = 0; k < FIN; k += 4) {
            v2f a, b;
            a.x = xr[k];
            a.y = xr[k + 1];
            b.x = wc[(size_t)k * HC];
            b.y = wc[(size_t)k * HC + HC];
            acc = __builtin_amdgcn_wmma_f32_16x16x4_f32(
                false, a, false, b, (short)0, acc, false, false);
        }
        // C/D layout: VGPR r, lanes 0-15 -> M=r; lanes 16-31 -> M=r+8; N=idx
        int srow = row0 + half * 8;
        int scol = col0 + idx;
        #pragma unroll
        for (int r = 0; r < 8; ++r)
            h[(size_t)(srow + r) * HC + scol] = acc[r];
    }
}

// ---------------- Kernel 2: per-(node,head) attention dots + init buffers ----------------
__global__ void gat_node_prep(const float* __restrict__ h,
                              const float* __restrict__ att_src,
                              const float* __restrict__ att_dst,
                              float* __restrict__ a_src,
                              float* __restrict__ a_dst,
                              float* __restrict__ emax,
                              float* __restrict__ denom,
                              float* __restrict__ agg,
                              int N) {
    int nh = blockIdx.x * blockDim.x + threadIdx.x;
    if (nh >= N * HEADS) return;
    int n  = nh / HEADS;
    int hd = nh - n * HEADS;
    const float* hrow = h + (size_t)n * HC + hd * CH;
    const float* as = att_src + hd * CH;
    const float* ad = att_dst + hd * CH;
    float* arow = agg + (size_t)n * HC + hd * CH;
    float sa = 0.f, sd = 0.f;
    #pragma unroll 8
    for (int c = 0; c < CH; ++c) {
        float v = hrow[c];
        sa = fmaf(v, as[c], sa);
        sd = fmaf(v, ad[c], sd);
        arow[c] = 0.f;              // zero aggregation buffer
    }
    a_src[nh] = sa;
    a_dst[nh] = sd;
    emax[nh]  = -INFINITY;
    denom[nh] = 0.f;
}

// ---------------- Kernel 3: segment max over incoming edges ----------------
__global__ void gat_edge_max(const int* __restrict__ src,
                             const int* __restrict__ dst,
                             const float* __restrict__ a_src,
                             const float* __restrict__ a_dst,
                             float* __restrict__ emax,
                             int E, int N) {
    int e = blockIdx.x * blockDim.x + threadIdx.x;
    int Etot = E + N;                        // self loops appended
    if (e >= Etot) return;
    int s, d;
    if (e < E) { s = src[e]; d = dst[e]; } else { s = d = e - E; }
    #pragma unroll
    for (int hd = 0; hd < HEADS; ++hd) {
        float v = lrelu(a_src[s * HEADS + hd] + a_dst[d * HEADS + hd]);
        atomicMax(&emax[d * HEADS + hd], v);
    }
}

// ---------------- Kernel 4: segment sum of exp ----------------
__global__ void gat_edge_expsum(const int* __restrict__ src,
                                const int* __restrict__ dst,
                                const float* __restrict__ a_src,
                                const float* __restrict__ a_dst,
                                const float* __restrict__ emax,
                                float* __restrict__ denom,
                                int E, int N) {
    int e = blockIdx.x * blockDim.x + threadIdx.x;
    int Etot = E + N;
    if (e >= Etot) return;
    int s, d;
    if (e < E) { s = src[e]; d = dst[e]; } else { s = d = e - E; }
    #pragma unroll
    for (int hd = 0; hd < HEADS; ++hd) {
        float v = lrelu(a_src[s * HEADS + hd] + a_dst[d * HEADS + hd]);
        atomicAdd(&denom[d * HEADS + hd], __expf(v - emax[d * HEADS + hd]));
    }
}

// ---------------- Kernel 5: weighted scatter-aggregate (wave32 per edge) ----------------
__global__ void gat_edge_aggregate(const int* __restrict__ src,
                                   const int* __restrict__ dst,
                                   const float* __restrict__ a_src,
                                   const float* __restrict__ a_dst,
                                   const float* __restrict__ emax,
                                   const float* __restrict__ denom,
                                   const float* __restrict__ h,
                                   float* __restrict__ agg,
                                   int E, int N) {
    int wave = (int)((blockIdx.x * blockDim.x + threadIdx.x) >> 5);
    int lane = threadIdx.x & 31;
    int Etot = E + N;
    if (wave >= Etot) return;
    int s, d;
    if (wave < E) { s = src[wave]; d = dst[wave]; } else { s = d = wave - E; }

    const float* hrow = h + (size_t)s * HC;
    __builtin_prefetch(hrow + lane, 0, 0);   // global_prefetch of source row (L2-resident)

    float alpha[HEADS];
    #pragma unroll
    for (int hd = 0; hd < HEADS; ++hd) {
        float v = lrelu(a_src[s * HEADS + hd] + a_dst[d * HEADS + hd]);
        alpha[hd] = __expf(v - emax[d * HEADS + hd]) /
                    (denom[d * HEADS + hd] + 1e-16f);
    }
    float* arow = agg + (size_t)d * HC;
    #pragma unroll
    for (int j = 0; j < HC / 32; ++j) {      // 6 coalesced channels per lane
        int c = lane + j * 32;
        atomicAdd(&arow[c], alpha[c >> 6] * hrow[c]);
    }
}

// ---------------- Kernel 6: out = relu(agg + bias1) @ W2 + b2 via WMMA ----------------
// N=40 padded to 48 (3 tiles of 16) with column guards.
__global__ void gat_gemm2_wmma(const float* __restrict__ agg,
                               const float* __restrict__ bias1,
                               const float* __restrict__ W2,
                               const float* __restrict__ b2,
                               float* __restrict__ out,
                               int n_tiles_total) {
    int wave = (int)((blockIdx.x * blockDim.x + threadIdx.x) >> 5);
    int lane = threadIdx.x & 31;
    if (wave < n_tiles_total) {
        const int NT = 3;                    // ceil(40/16)
        int mt = wave / NT;
        int nt = wave - mt * NT;
        int row0 = mt * 16, col0 = nt * 16;
        int idx  = lane & 15;
        int half = lane >> 4;
        int arow = row0 + idx;
        int bcol = col0 + idx;
        bool bvalid = bcol < OUTF;

        v8f acc = {};
        #pragma unroll 4
        for (int k = 0; k < HC; k += 4) {
            int k0 = k + 2 * half;
            v2f a, b;
            float a0 = agg[(size_t)arow * HC + k0]     + bias1[k0];
            float a1 = agg[(size_t)arow * HC + k0 + 1] + bias1[k0 + 1];
            a.x = a0 > 0.f ? a0 : 0.f;       // relu epilogue of stage 1 fused into A
            a.y = a1 > 0.f ? a1 : 0.f;
            b.x = bvalid ? W2[(size_t)k0 * OUTF + bcol]       : 0.f;
            b.y = bvalid ? W2[(size_t)(k0 + 1) * OUTF + bcol] : 0.f;
            acc = __builtin_amdgcn_wmma_f32_16x16x4_f32(
                false, a, false, b, (short)0, acc, false, false);
        }
        int srow = row0 + half * 8;
        int scol = col0 + idx;
        if (scol < OUTF) {
            float bb = b2[scol];
            #pragma unroll
            for (int r = 0; r < 8; ++r)
                out[(size_t)(srow + r) * OUTF + scol] = acc[r] + bb;
        }
    }
}

extern "C" void kernel_launch(void* const* d_in, const int* in_sizes, int n_in,
                              void* d_out, int out_size, void* d_ws, size_t ws_size,
                              hipStream_t stream) {
    const float* x       = (const float*)d_in[0];
    const float* W1      = (const float*)d_in[1];
    const float* att_src = (const float*)d_in[2];
    const float* att_dst = (const float*)d_in[3];
    const float* bias1   = (const float*)d_in[4];
    const float* W2      = (const float*)d_in[5];
    const float* b2      = (const float*)d_in[6];
    const int*   ei      = (const int*)d_in[7];

    int N = in_sizes[0] / FIN;       // 50000
    int E = in_sizes[7] / 2;         // 1000000
    const int* src = ei;
    const int* dst = ei + E;
    float* out = (float*)d_out;

    // workspace layout
    float* h     = (float*)d_ws;
    float* agg   = h     + (size_t)N * HC;
    float* a_src = agg   + (size_t)N * HC;
    float* a_dst = a_src + (size_t)N * HEADS;
    float* emax  = a_dst + (size_t)N * HEADS;
    float* denom = emax  + (size_t)N * HEADS;

    // K1: h = x @ W1   (tiles: (N/16) x 12, one wave per tile)
    {
        int tiles = (N / 16) * (HC / 16);
        int threads = tiles * 32;
        int block = 256;
        int grid = (threads + block - 1) / block;
        gat_gemm1_wmma<<<grid, block, 0, stream>>>(x, W1, h, tiles, 0);
    }
    // K2: attention dots + buffer init
    {
        int threads = N * HEADS;
        int block = 256;
        int grid = (threads + block - 1) / block;
        gat_node_prep<<<grid, block, 0, stream>>>(h, att_src, att_dst,
                                                  a_src, a_dst, emax, denom, agg, N);
    }
    // K3: segment max
    {
        int threads = E + N;
        int block = 256;
        int grid = (threads + block - 1) / block;
        gat_edge_max<<<grid, block, 0, stream>>>(src, dst, a_src, a_dst, emax, E, N);
    }
    // K4: segment exp-sum
    {
        int threads = E + N;
        int block = 256;
        int grid = (threads + block - 1) / block;
        gat_edge_expsum<<<grid, block, 0, stream>>>(src, dst, a_src, a_dst,
                                                    emax, denom, E, N);
    }
    // K5: weighted scatter aggregation (one wave32 per edge)
    {
        long long threads = (long long)(E + N) * 32;
        int block = 256;
        long long grid = (threads + block - 1) / block;
        gat_edge_aggregate<<<(int)grid, block, 0, stream>>>(src, dst, a_src, a_dst,
                                                            emax, denom, h, agg, E, N);
    }
    // K6: out = relu(agg + bias1) @ W2 + b2
    {
        int tiles = (N / 16) * 3;
        int threads = tiles * 32;
        int block = 256;
        int grid = (threads + block - 1) / block;
        gat_gemm2_wmma<<<grid, block, 0, stream>>>(agg, bias1, W2, b2, out, tiles);
    }
}